// VectorQuantizer_50783693308504
// MI455X (gfx1250) — compile-verified
//
#include <hip/hip_runtime.h>
#include <hip/hip_bf16.h>
#include <stdint.h>
#include <stddef.h>

// ---------------------------------------------------------------------------
// VQ-VAE vector quantizer for MI455X (gfx1250, wave32).
//   scores = z @ codebook^T  (32768 x 8192, inner D=512, f32) via
//   V_WMMA_F32_16X16X4_F32; argmin(dist) == argmax(x.e - 0.5*|e|^2).
// ~275 GFLOP f32 matrix work dominates (~6us of mandatory HBM traffic at
// 23.3 TB/s); the 16MB codebook is L2-resident (192MB L2), so every block
// streams all 8192 codes from L2. Tile staging uses the gfx1250 async
// global->LDS path (GLOBAL_LOAD_ASYNC_TO_LDS_B128, ASYNCcnt-tracked).
// ---------------------------------------------------------------------------

typedef float v2f __attribute__((ext_vector_type(2)));
typedef float v8f __attribute__((ext_vector_type(8)));

#define N_ROWS     32768          // 8 * 4096
#define K_CODES    8192
#define D_DIM      512
#define BM         128            // z-rows per block (8 waves x 16 rows)
#define BN         32             // codes per chunk (2 col tiles per wave)
#define BK         64             // k-slice staged in LDS
#define LDS_STRIDE 68             // floats; 272B row: 16B aligned + bank skew
#define NUM_CHUNKS (K_CODES / BN) // 256
#define NUM_KC     (D_DIM / BK)   // 8

// d_out layout (flat, return order): z_q_ste | vq_loss | indices | perplexity
#define ZQ_ELEMS   (N_ROWS * D_DIM)            // 16777216
#define VQ_OFF     (ZQ_ELEMS)
#define IDX_OFF    (ZQ_ELEMS + 1)
#define PERP_OFF   (ZQ_ELEMS + 1 + N_ROWS)

// ---- gfx1250 async global->LDS copy (B128), ASYNCcnt-tracked --------------
// Builtin signature (from clang diagnostic): both pointer params are
// `int __attribute__((vector_size(16)))*`, src in AS(1), dst in AS(3).
#if defined(__has_builtin)
#if __has_builtin(__builtin_amdgcn_global_load_async_to_lds_b128)
#define HAVE_ASYNC_LDS 1
#endif
#endif

#ifdef HAVE_ASYNC_LDS
typedef int v4i __attribute__((vector_size(16)));
typedef __attribute__((address_space(1))) v4i gv4i;
typedef __attribute__((address_space(3))) v4i lv4i;
#define ASYNC_COPY_B128(gptr, lptr)                                          \
  __builtin_amdgcn_global_load_async_to_lds_b128((gv4i*)(gptr),              \
                                                 (lv4i*)(lptr), 0, 0)
#if __has_builtin(__builtin_amdgcn_s_wait_asynccnt)
#define ASYNC_WAIT0() __builtin_amdgcn_s_wait_asynccnt(0)
#else
#define ASYNC_WAIT0() asm volatile("s_wait_asynccnt 0x0" ::: "memory")
#endif
#endif

// ---------------------------------------------------------------------------
__global__ void vq_zero_kernel(int* __restrict__ counts) {
  int i = blockIdx.x * blockDim.x + threadIdx.x;
  if (i < K_CODES) counts[i] = 0;
}

// 0.5 * |e_k|^2 per code, one wave per codebook row.
__global__ __launch_bounds__(256) void vq_e2_kernel(const float* __restrict__ cb,
                                                    float* __restrict__ e2h) {
  const int wave = threadIdx.x >> 5, lane = threadIdx.x & 31;
  const int code = blockIdx.x * 8 + wave;
  const float4* row = (const float4*)(cb + (size_t)code * D_DIM);
  float s = 0.f;
#pragma unroll
  for (int j = 0; j < 4; ++j) {
    float4 c = row[lane + j * 32];
    s += c.x * c.x + c.y * c.y + c.z * c.z + c.w * c.w;
  }
#pragma unroll
  for (int m = 16; m >= 1; m >>= 1) s += __shfl_xor(s, m, 32);
  if (lane == 0) e2h[code] = 0.5f * s;
}

// ---------------------------------------------------------------------------
// Argmin search: block = 256 threads = 8 waves. Each wave owns a 16-row tile
// and computes a 16x32 C tile per chunk (two accumulators -> A-frag reuse x2,
// two independent WMMA chains). Block: 128 rows x all 8192 codes.
__global__ __launch_bounds__(256) void vq_argmin_kernel(
    const float* __restrict__ z, const float* __restrict__ cb,
    const float* __restrict__ e2h, int* __restrict__ idx_out) {
  __shared__ float Atile[BM * LDS_STRIDE];   // 34.0 KB
  __shared__ float Btile[BN * LDS_STRIDE];   //  8.5 KB

  const int tid     = threadIdx.x;
  const int lane    = tid & 31;
  const int wave    = tid >> 5;              // 0..7: 16-row tile owner
  const int rowBase = blockIdx.x * BM;

  // WMMA f32 16x16x4 fragment addressing (wave32):
  //  A: lane L holds row M=L&15, K = 2*(L>>4) + {0,1}  -> one b64 LDS load
  //  B[K][N] = codebook[N][K] -> same pattern on codebook rows.
  const int fr    = lane & 15;
  const int khalf = (lane >> 4) << 1;        // 0 or 2
  const float* Afrag  = &Atile[(wave * 16 + fr) * LDS_STRIDE + khalf];
  const float* Bfrag0 = &Btile[fr * LDS_STRIDE + khalf];
  const float* Bfrag1 = &Btile[(16 + fr) * LDS_STRIDE + khalf];

  float bestv[8];
  int   besti[8];
#pragma unroll
  for (int v = 0; v < 8; ++v) { bestv[v] = -3.4e38f; besti[v] = 0; }

  for (int chunk = 0; chunk < NUM_CHUNKS; ++chunk) {
    const int codeBase = chunk * BN;

    // Hint next chunk's codebook rows toward the caches (global_prefetch_b8).
    if (chunk + 1 < NUM_CHUNKS) {
      const float* p =
          cb + (size_t)(codeBase + BN + (tid >> 3)) * D_DIM + (tid & 7) * 64;
      __builtin_prefetch(p, 0, 1);
    }

    v8f acc0 = {0.f, 0.f, 0.f, 0.f, 0.f, 0.f, 0.f, 0.f};
    v8f acc1 = {0.f, 0.f, 0.f, 0.f, 0.f, 0.f, 0.f, 0.f};

    for (int kc = 0; kc < NUM_KC; ++kc) {
      const int kBase = kc * BK;
      __syncthreads();  // all waves done reading LDS before restaging
#ifdef HAVE_ASYNC_LDS
      // A tile: 128x64 f32 = 2048 x b128; 8 per thread, direct to LDS.
#pragma unroll
      for (int j = 0; j < 8; ++j) {
        int i = tid + j * 256;
        int r = i >> 4, c4 = (i & 15) << 2;
        ASYNC_COPY_B128(z + (size_t)(rowBase + r) * D_DIM + kBase + c4,
                        &Atile[r * LDS_STRIDE + c4]);
      }
      // B tile: 32x64 f32 = 512 x b128; 2 per thread.
#pragma unroll
      for (int j = 0; j < 2; ++j) {
        int i = tid + j * 256;
        int r = i >> 4, c4 = (i & 15) << 2;
        ASYNC_COPY_B128(cb + (size_t)(codeBase + r) * D_DIM + kBase + c4,
                        &Btile[r * LDS_STRIDE + c4]);
      }
      ASYNC_WAIT0();
#else
#pragma unroll
      for (int j = 0; j < 8; ++j) {
        int i = tid + j * 256;
        int r = i >> 4, c4 = (i & 15) << 2;
        float4 val = *(const float4*)(z + (size_t)(rowBase + r) * D_DIM + kBase + c4);
        *(float4*)(&Atile[r * LDS_STRIDE + c4]) = val;
      }
#pragma unroll
      for (int j = 0; j < 2; ++j) {
        int i = tid + j * 256;
        int r = i >> 4, c4 = (i & 15) << 2;
        float4 val = *(const float4*)(cb + (size_t)(codeBase + r) * D_DIM + kBase + c4);
        *(float4*)(&Btile[r * LDS_STRIDE + c4]) = val;
      }
#endif
      __syncthreads();

      // 16 WMMA steps x 2 col tiles cover the 64-wide k-slice (K=4 each).
#pragma unroll
      for (int ks = 0; ks < 16; ++ks) {
        v2f a  = *(const v2f*)(Afrag + ks * 4);
        v2f b0 = *(const v2f*)(Bfrag0 + ks * 4);
        v2f b1 = *(const v2f*)(Bfrag1 + ks * 4);
        acc0 = __builtin_amdgcn_wmma_f32_16x16x4_f32(
            false, a, false, b0, (short)0, acc0, false, false);
        acc1 = __builtin_amdgcn_wmma_f32_16x16x4_f32(
            false, a, false, b1, (short)0, acc1, false, false);
      }
    }

    // Epilogue: score = x.e - 0.5|e|^2; running argmax (== distance argmin).
    // Col tile 0 (smaller codes) processed first; strict '>' keeps first min.
    const int   code0 = codeBase + fr;
    const int   code1 = code0 + 16;
    const float e20   = e2h[code0];
    const float e21   = e2h[code1];
#pragma unroll
    for (int v = 0; v < 8; ++v) {
      float c0 = acc0[v] - e20;
      if (c0 > bestv[v]) { bestv[v] = c0; besti[v] = code0; }
      float c1 = acc1[v] - e21;
      if (c1 > bestv[v]) { bestv[v] = c1; besti[v] = code1; }
    }
  }

  // Reduce across the 16 lanes of each half-wave (C layout: lanes 0-15 = rows
  // v, lanes 16-31 = rows v+8). Ties prefer the smaller code index.
#pragma unroll
  for (int v = 0; v < 8; ++v) {
#pragma unroll
    for (int m = 8; m >= 1; m >>= 1) {
      float ov = __shfl_xor(bestv[v], m, 32);
      int   oi = __shfl_xor(besti[v], m, 32);
      if (ov > bestv[v] || (ov == bestv[v] && oi < besti[v])) {
        bestv[v] = ov; besti[v] = oi;
      }
    }
  }
  // Each row is owned by exactly one wave: write indices directly.
  if (lane == 0) {
#pragma unroll
    for (int v = 0; v < 8; ++v) idx_out[rowBase + wave * 16 + v] = besti[v];
  } else if (lane == 16) {
#pragma unroll
    for (int v = 0; v < 8; ++v) idx_out[rowBase + wave * 16 + 8 + v] = besti[v];
  }
}

// ---------------------------------------------------------------------------
// Gather z_q (== z_q_ste value), deterministic SSE partials, histogram.
__global__ __launch_bounds__(256) void vq_gather_kernel(
    const float* __restrict__ z, const float* __restrict__ cb,
    const int* __restrict__ idx, float* __restrict__ zq,
    int* __restrict__ counts, float* __restrict__ partials) {
  __shared__ float wsse[8];
  const int wave = threadIdx.x >> 5, lane = threadIdx.x & 31;
  const int row  = blockIdx.x * 8 + wave;
  const int code = idx[row];
  const float4* cr   = (const float4*)(cb + (size_t)code * D_DIM);
  const float4* zr   = (const float4*)(z + (size_t)row * D_DIM);
  float4*       outr = (float4*)(zq + (size_t)row * D_DIM);
  float s = 0.f;
#pragma unroll
  for (int j = 0; j < 4; ++j) {
    int i = lane + j * 32;
    float4 c = cr[i], zz = zr[i];
    outr[i] = c;
    float dx = c.x - zz.x, dy = c.y - zz.y, dz = c.z - zz.z, dw = c.w - zz.w;
    s += dx * dx + dy * dy + dz * dz + dw * dw;
  }
#pragma unroll
  for (int m = 16; m >= 1; m >>= 1) s += __shfl_xor(s, m, 32);
  if (lane == 0) {
    wsse[wave] = s;
    atomicAdd(&counts[code], 1);   // integer atomics: order-independent
  }
  __syncthreads();
  if (threadIdx.x == 0) {
    float p = 0.f;
#pragma unroll
    for (int w = 0; w < 8; ++w) p += wsse[w];  // fixed order -> deterministic
    partials[blockIdx.x] = p;
  }
}

// ---------------------------------------------------------------------------
// vq_loss = 1.25 * SSE / numel ; perplexity = exp(-sum p*log(p+1e-10)).
__global__ __launch_bounds__(256) void vq_finalize_kernel(
    const float* __restrict__ partials, const int* __restrict__ counts,
    float* __restrict__ out) {
  __shared__ float red[256];
  const int t = threadIdx.x;
  float s = 0.f;
  for (int j = 0; j < (N_ROWS / 8); j += 256) s += partials[t + j];
  red[t] = s;
  __syncthreads();
  for (int off = 128; off > 0; off >>= 1) {
    if (t < off) red[t] += red[t + off];
    __syncthreads();
  }
  const float sse = red[0];
  __syncthreads();
  float h = 0.f;
  for (int k = t; k < K_CODES; k += 256) {
    float p = (float)counts[k] * (1.0f / (float)N_ROWS);
    h += p * logf(p + 1e-10f);
  }
  red[t] = h;
  __syncthreads();
  for (int off = 128; off > 0; off >>= 1) {
    if (t < off) red[t] += red[t + off];
    __syncthreads();
  }
  if (t == 0) {
    out[VQ_OFF]   = 1.25f * sse * (1.0f / (float)ZQ_ELEMS);
    out[PERP_OFF] = expf(-red[0]);
  }
}

// ---------------------------------------------------------------------------
extern "C" void kernel_launch(void* const* d_in, const int* in_sizes, int n_in,
                              void* d_out, int out_size, void* d_ws, size_t ws_size,
                              hipStream_t stream) {
  const float* z  = (const float*)d_in[0];   // (8,4096,512) f32
  const float* cb = (const float*)d_in[1];   // (8192,512)   f32
  float* out      = (float*)d_out;

  // workspace: e2h[8192] f32 | counts[8192] i32 | partials[4096] f32  (~80KB)
  float* e2h      = (float*)d_ws;
  int*   counts   = (int*)((char*)d_ws + (size_t)K_CODES * sizeof(float));
  float* partials = (float*)((char*)d_ws + (size_t)2 * K_CODES * sizeof(float));
  int*   idx_out  = (int*)out + IDX_OFF;

  vq_zero_kernel<<<K_CODES / 256, 256, 0, stream>>>(counts);
  vq_e2_kernel<<<K_CODES / 8, 256, 0, stream>>>(cb, e2h);
  vq_argmin_kernel<<<N_ROWS / BM, 256, 0, stream>>>(z, cb, e2h, idx_out);
  vq_gather_kernel<<<N_ROWS / 8, 256, 0, stream>>>(z, cb, idx_out, out, counts, partials);
  vq_finalize_kernel<<<1, 256, 0, stream>>>(partials, counts, out);
}